// SimpleGAT_78743930405200
// MI455X (gfx1250) — compile-verified
//
#include <hip/hip_runtime.h>
#include <hip/hip_bf16.h>

#define N_NODES 50000
#define N_EDGES 800000
#define F 64
#define SLOPE 0.2f

typedef __attribute__((ext_vector_type(2))) float v2f;
typedef __attribute__((ext_vector_type(8))) float v8f;

__device__ __forceinline__ float leaky(float x) { return x > 0.f ? x : SLOPE * x; }

// Float atomic max via sign-aware integer atomics (stores real float bit patterns).
__device__ __forceinline__ void atomicMaxFloat(float* addr, float v) {
  if (v >= 0.f) atomicMax((int*)addr, __float_as_int(v));
  else          atomicMin((unsigned int*)addr, __float_as_uint(v));
}

// ---------------------------------------------------------------------------
// Init: zero output accumulator, denom; seg_max = -inf.
// ---------------------------------------------------------------------------
__global__ void gat_init(float* __restrict__ out, float* __restrict__ segm,
                         float* __restrict__ den) {
  int i = blockIdx.x * blockDim.x + threadIdx.x;
  if (i < N_NODES * F) out[i] = 0.f;
  if (i < N_NODES) { segm[i] = -__builtin_inff(); den[i] = 0.f; }
}

// ---------------------------------------------------------------------------
// Node transform: h = leaky_relu(X @ W^T + b_w) via V_WMMA_F32_16X16X4_F32.
// One wave computes a 16-node x 64-feature tile (4 accumulators, 16 K-steps).
// A frag (16x4 f32): lane L holds X[tile*16 + (L&15)][4s + 2*(L>>4) + {0,1}]
// B frag (4x16 f32): B[k][n] = W[n][k]; lane L holds W[n0+(L&15)][4s+2*(L>>4)+{0,1}]
// C/D (16x16 f32):   VGPR r, lane L -> row r + 8*(L>>4), col n0 + (L&15)
// ---------------------------------------------------------------------------
__global__ __launch_bounds__(128) void gat_node_gemm(
    const float* __restrict__ X, const float* __restrict__ Wm,
    const float* __restrict__ bw, float* __restrict__ h) {
  const int wave = threadIdx.x >> 5;
  const int tile = blockIdx.x * 4 + wave;
  if (tile >= N_NODES / 16) return;            // uniform per-wave exit (EXEC all-1 for WMMA)
  const int lane = threadIdx.x & 31;
  const int half = lane >> 4;                  // 0: K={0,1}, 1: K={2,3} within a step
  const int l16  = lane & 15;

  const float* arow = X + (size_t)(tile * 16 + l16) * F + 2 * half;

  v8f acc[4] = {};
#pragma unroll
  for (int s = 0; s < 16; ++s) {
    v2f af = *(const v2f*)(arow + 4 * s);
#pragma unroll
    for (int nt = 0; nt < 4; ++nt) {
      v2f bf = *(const v2f*)(Wm + (size_t)(nt * 16 + l16) * F + 4 * s + 2 * half);
      acc[nt] = __builtin_amdgcn_wmma_f32_16x16x4_f32(
          false, af, false, bf, (short)0, acc[nt], false, false);
    }
  }

#pragma unroll
  for (int nt = 0; nt < 4; ++nt) {
    const int n = nt * 16 + l16;
    const float bias = bw[n];
#pragma unroll
    for (int r = 0; r < 8; ++r) {
      const int m = tile * 16 + r + 8 * half;
      h[(size_t)m * F + n] = leaky(acc[nt][r] + bias);
    }
  }
}

// ---------------------------------------------------------------------------
// Per-node attention scalars: s1[n] = h[n].a1, s2[n] = h[n].a2
// (turns the per-edge 128-FLOP dot products into 2 scalar gathers per edge)
// ---------------------------------------------------------------------------
__global__ void gat_node_scores(const float* __restrict__ h,
                                const float* __restrict__ a,
                                float* __restrict__ s1, float* __restrict__ s2) {
  int n = blockIdx.x * blockDim.x + threadIdx.x;
  if (n >= N_NODES) return;
  const float4* hr = (const float4*)(h + (size_t)n * F);
  const float4* a1 = (const float4*)a;
  const float4* a2 = (const float4*)(a + F);
  float t1 = 0.f, t2 = 0.f;
#pragma unroll
  for (int k = 0; k < F / 4; ++k) {
    float4 hv = hr[k], x1 = a1[k], x2 = a2[k];
    t1 += hv.x * x1.x + hv.y * x1.y + hv.z * x1.z + hv.w * x1.w;
    t2 += hv.x * x2.x + hv.y * x2.y + hv.z * x2.z + hv.w * x2.w;
  }
  s1[n] = t1; s2[n] = t2;
}

// ---------------------------------------------------------------------------
// Edge logits + segment max
// ---------------------------------------------------------------------------
__global__ void gat_edge_logits(const int* __restrict__ tgt, const int* __restrict__ nbr,
                                const float* __restrict__ s1, const float* __restrict__ s2,
                                const float* __restrict__ ba,
                                float* __restrict__ eij, float* __restrict__ segm) {
  int e = blockIdx.x * blockDim.x + threadIdx.x;
  if (e >= N_EDGES) return;
  int t = tgt[e], nb = nbr[e];
  float v = s1[t] + s2[nb] + ba[0];
  eij[e] = v;
  atomicMaxFloat(segm + t, v);
}

// ---------------------------------------------------------------------------
// exp(eij - max) + segment sum (eij buffer reused to hold e_exp)
// ---------------------------------------------------------------------------
__global__ void gat_edge_expsum(const int* __restrict__ tgt,
                                float* __restrict__ eij,
                                const float* __restrict__ segm,
                                float* __restrict__ den) {
  int e = blockIdx.x * blockDim.x + threadIdx.x;
  if (e >= N_EDGES) return;
  int t = tgt[e];
  float ee = __expf(eij[e] - segm[t]);
  eij[e] = ee;
  atomicAdd(den + t, ee);
}

// ---------------------------------------------------------------------------
// Weighted message scatter: 64 threads per edge, coalesced h-row gather and
// coalesced global_atomic_add_f32 into the output accumulator.
// ---------------------------------------------------------------------------
__global__ __launch_bounds__(256) void gat_scatter(
    const int* __restrict__ tgt, const int* __restrict__ nbr,
    const float* __restrict__ eexp, const float* __restrict__ den,
    const float* __restrict__ h, float* __restrict__ out) {
  int tix = threadIdx.x;
  int e = blockIdx.x * 4 + (tix >> 6);
  int f = tix & 63;
  if (e >= N_EDGES) return;
  int t = tgt[e], nb = nbr[e];
  float alpha = eexp[e] / den[t];
  atomicAdd(out + (size_t)t * F + f, alpha * h[(size_t)nb * F + f]);
}

// ---------------------------------------------------------------------------
// Final leaky_relu in place on the accumulator (= d_out)
// ---------------------------------------------------------------------------
__global__ void gat_finalize(float* __restrict__ out) {
  int i = blockIdx.x * blockDim.x + threadIdx.x;
  if (i < N_NODES * F) out[i] = leaky(out[i]);
}

extern "C" void kernel_launch(void* const* d_in, const int* in_sizes, int n_in,
                              void* d_out, int out_size, void* d_ws, size_t ws_size,
                              hipStream_t stream) {
  const float* X  = (const float*)d_in[0];              // [50000,64]
  const int*   ei = (const int*)d_in[1];                // [2,800000] (JAX default int32)
  const float* Wm = (const float*)d_in[2];              // [64,64]
  const float* bw = (const float*)d_in[3];              // [64]
  const float* a  = (const float*)d_in[4];              // [1,128]
  const float* ba = (const float*)d_in[5];              // [1]
  const int* tgt = ei;
  const int* nbr = ei + N_EDGES;

  float* out = (float*)d_out;

  float* ws   = (float*)d_ws;
  float* h    = ws;                                     // 3,200,000 f32
  float* s1   = h + (size_t)N_NODES * F;                //    50,000
  float* s2   = s1 + N_NODES;                           //    50,000
  float* segm = s2 + N_NODES;                           //    50,000
  float* den  = segm + N_NODES;                         //    50,000
  float* eij  = den + N_NODES;                          //   800,000 (logits, then e_exp)

  gat_init<<<(N_NODES * F + 255) / 256, 256, 0, stream>>>(out, segm, den);
  gat_node_gemm<<<(N_NODES / 16 + 3) / 4, 128, 0, stream>>>(X, Wm, bw, h);
  gat_node_scores<<<(N_NODES + 255) / 256, 256, 0, stream>>>(h, a, s1, s2);
  gat_edge_logits<<<(N_EDGES + 255) / 256, 256, 0, stream>>>(tgt, nbr, s1, s2, ba, eij, segm);
  gat_edge_expsum<<<(N_EDGES + 255) / 256, 256, 0, stream>>>(tgt, eij, segm, den);
  gat_scatter<<<N_EDGES / 4, 256, 0, stream>>>(tgt, nbr, eij, den, h, out);
  gat_finalize<<<(N_NODES * F + 255) / 256, 256, 0, stream>>>(out);
}